// RelationalGraphConv_5669356834165
// MI455X (gfx1250) — compile-verified
//
#include <hip/hip_runtime.h>

#define N_NODES 50000
#define DIM     128
#define NREL    8
#define NEDGE   800000
#define NSEG    (N_NODES * NREL)   // 400000
#define EPS     1e-10f

typedef __attribute__((ext_vector_type(2))) float v2f;
typedef __attribute__((ext_vector_type(8))) float v8f;

// ---------------------------------------------------------------------------
// Kernel 1: zero num[NSEG*DIM] + den[NSEG] (contiguous in d_ws)
// ---------------------------------------------------------------------------
__global__ __launch_bounds__(256) void rgcn_zero_ws(float4* __restrict__ p, long n4) {
    long i = (long)blockIdx.x * blockDim.x + threadIdx.x;
    long stride = (long)gridDim.x * blockDim.x;
    float4 z = make_float4(0.f, 0.f, 0.f, 0.f);
    for (; i < n4; i += stride) p[i] = z;
}

// ---------------------------------------------------------------------------
// Kernel 2: edge scatter. One thread per (edge, feature dim).
//   num[seg][d] += x[src][d] * w ;  den[seg] += w (lane d==0)
// ---------------------------------------------------------------------------
__global__ __launch_bounds__(256) void rgcn_scatter(
    const float* __restrict__ x, const float* __restrict__ ew,
    const int* __restrict__ nin, const int* __restrict__ nout,
    const int* __restrict__ rel,
    float* __restrict__ num, float* __restrict__ den) {
    long t = (long)blockIdx.x * blockDim.x + threadIdx.x;
    if (t >= (long)NEDGE * DIM) return;
    int e = (int)(t >> 7);
    int d = (int)(t & (DIM - 1));
    int src = nin[e];
    float w  = ew[e];
    int seg  = nout[e] * NREL + rel[e];
    atomicAdd(&num[(long)seg * DIM + d], x[(long)src * DIM + d] * w);
    if (d == 0) atomicAdd(&den[seg], w);
}

// ---------------------------------------------------------------------------
// Kernel 3: fused GEMM
//   out[n, :] = relu( [num/den | x](n, 0:1152) @ [Wlin; Wloop] + blin + bloop )
// One block = 8 waves = 16 output rows x 128 output cols (one 16x16 tile/wave).
// A panel staged in LDS (padded stride), B read from global (L2 resident).
// ---------------------------------------------------------------------------
#define KA   1152              // 1024 (relations) + 128 (self loop)
#define AS   1156              // padded LDS row stride (mod 64 == 4 banks)

__global__ __launch_bounds__(256) void rgcn_gemm(
    const float* __restrict__ num, const float* __restrict__ den,
    const float* __restrict__ x,
    const float* __restrict__ Wlin,  const float* __restrict__ blin,
    const float* __restrict__ Wloop, const float* __restrict__ bloop,
    float* __restrict__ out) {
    __shared__ __align__(16) float ldsA[16 * AS];
    __shared__ float dinv[16 * NREL];

    const int tid     = threadIdx.x;
    const int rowBase = blockIdx.x * 16;          // 3125 blocks * 16 = 50000 exactly

    // 1 / (den + eps) for the 16x8 (row, relation) slots of this panel
    if (tid < 16 * NREL) {
        dinv[tid] = 1.0f / (den[(long)rowBase * NREL + tid] + EPS);
    }
    __syncthreads();

    // Stage A = [update rows | x rows] into LDS, scaling by 1/den on the fly
    for (int idx = tid; idx < 16 * KA; idx += 256) {
        int row = idx / KA;
        int col = idx - row * KA;
        float v;
        if (col < NREL * DIM) {
            int r = col >> 7;            // relation
            int d = col & (DIM - 1);
            v = num[((long)(rowBase + row) * NREL + r) * DIM + d] * dinv[row * NREL + r];
        } else {
            v = x[(long)(rowBase + row) * DIM + (col - NREL * DIM)];
        }
        ldsA[row * AS + col] = v;
    }
    __syncthreads();

    // Per-lane fragment coordinates (ISA 16x16x4 f32 layout)
    const int lane = tid & 31;
    const int wave = tid >> 5;
    const int colL = lane & 15;                    // tile column
    const int kb   = (lane >> 4) << 1;             // K sub-offset: 0 or 2
    const int colG = wave * 16 + colL;             // global output column (0..127)
    const float* arow = &ldsA[(lane & 15) * AS];   // A row for this lane

    v8f c = {0.f, 0.f, 0.f, 0.f, 0.f, 0.f, 0.f, 0.f};

    // Phase 1: K = 0..1023 against W_lin [1024 x 128]
#pragma unroll 4
    for (int k = 0; k < NREL * DIM; k += 4) {
        v2f a = *(const v2f*)(&arow[k + kb]);
        v2f b;
        b.x = Wlin[(k + kb)     * DIM + colG];
        b.y = Wlin[(k + kb + 1) * DIM + colG];
        c = __builtin_amdgcn_wmma_f32_16x16x4_f32(false, a, false, b,
                                                  (short)0, c, false, false);
    }
    // Phase 2: K = 0..127 against W_loop [128 x 128] (A offset 1024)
#pragma unroll 4
    for (int k = 0; k < DIM; k += 4) {
        v2f a = *(const v2f*)(&arow[NREL * DIM + k + kb]);
        v2f b;
        b.x = Wloop[(k + kb)     * DIM + colG];
        b.y = Wloop[(k + kb + 1) * DIM + colG];
        c = __builtin_amdgcn_wmma_f32_16x16x4_f32(false, a, false, b,
                                                  (short)0, c, false, false);
    }

    // Epilogue: bias + relu + store.  C/D layout: VGPR i -> row i (lanes 0-15),
    // row i+8 (lanes 16-31); column = lane & 15.
    const float bias = blin[colG] + bloop[colG];
    const int rowOff = (lane >> 4) * 8;
#pragma unroll
    for (int i = 0; i < 8; ++i) {
        float v = c[i] + bias;
        v = v > 0.f ? v : 0.f;
        out[(long)(rowBase + rowOff + i) * DIM + colG] = v;
    }
}

// ---------------------------------------------------------------------------
extern "C" void kernel_launch(void* const* d_in, const int* in_sizes, int n_in,
                              void* d_out, int out_size, void* d_ws, size_t ws_size,
                              hipStream_t stream) {
    const float* x     = (const float*)d_in[0];
    const float* ew    = (const float*)d_in[1];
    const float* Wlin  = (const float*)d_in[2];
    const float* blin  = (const float*)d_in[3];
    const float* Wloop = (const float*)d_in[4];
    const float* bloop = (const float*)d_in[5];
    const int*   nin   = (const int*)d_in[6];
    const int*   nout  = (const int*)d_in[7];
    const int*   rel   = (const int*)d_in[8];
    float* out = (float*)d_out;

    float* num = (float*)d_ws;                       // [NSEG, DIM]
    float* den = num + (long)NSEG * DIM;             // [NSEG]

    // Zero workspace (num + den are contiguous; total floats divisible by 4)
    long totalFloats = (long)NSEG * DIM + NSEG;      // 51,600,000
    rgcn_zero_ws<<<2048, 256, 0, stream>>>((float4*)d_ws, totalFloats / 4);

    // Edge scatter: one thread per (edge, dim)
    long scatterThreads = (long)NEDGE * DIM;         // 102,400,000
    int scatterBlocks = (int)((scatterThreads + 255) / 256);
    rgcn_scatter<<<scatterBlocks, 256, 0, stream>>>(x, ew, nin, nout, rel, num, den);

    // Fused WMMA GEMM: 16 rows per block, 50000/16 = 3125 blocks
    rgcn_gemm<<<N_NODES / 16, 256, 0, stream>>>(num, den, x, Wlin, blin,
                                                Wloop, bloop, out);
}